// GATLayerEdgeSoftmax_40827959116587
// MI455X (gfx1250) — compile-verified
//
#include <hip/hip_runtime.h>
#include <math.h>

typedef float v2f __attribute__((ext_vector_type(2)));
typedef float v8f __attribute__((ext_vector_type(8)));

#define D 256
#define NEG_SLOPE 0.01f
#define EPS_F 1e-6f

__device__ __forceinline__ float lrelu(float x) {
    return x >= 0.0f ? x : NEG_SLOPE * x;
}

// ---------------------------------------------------------------------------
// Kernel 1: zero segment-sum buffer (workspace is poisoned, must init per call)
// ---------------------------------------------------------------------------
__global__ void gat_zero_kernel(float* __restrict__ S, int n) {
    int i = blockIdx.x * blockDim.x + threadIdx.x;
    if (i < n) S[i] = 0.0f;
}

// ---------------------------------------------------------------------------
// Kernel 2: fused node-level GEMMs with fp32 WMMA (16x16x4).
//   U = X @ W1^T            (W1 = W_in[:, :256])
//   V = X @ W2^T + b_in     (W2 = W_in[:, 256:])
//   Y = X @ Wout^T + b_out
// One block per 16-node tile. X tile staged in LDS. 8 waves x 6 col-tiles
// cover the 48 output 16-col tiles (16 each for U, V, Y).
// A-frag layout (ISA 7.12.2, 32-bit A 16x4): lanes 0-15: {K, K+1},
// lanes 16-31: {K+2, K+3}; row M = lane&15. B-frag mirrored (N = lane&15).
// ---------------------------------------------------------------------------
__global__ __launch_bounds__(256)
void gat_node_gemm_kernel(const float* __restrict__ X,
                          const float* __restrict__ W_in,
                          const float* __restrict__ b_in,
                          const float* __restrict__ W_out,
                          const float* __restrict__ b_out,
                          float* __restrict__ U,
                          float* __restrict__ V,
                          float* __restrict__ Y,
                          int N) {
    __shared__ float Xs[16][D + 4];   // +4 pad: conflict-free strided reads
    const int node0 = blockIdx.x * 16;
    const int t = threadIdx.x;

    // Cooperative LDS fill: each thread moves 16 consecutive floats (4x b128).
    {
        int row  = t >> 4;
        int col  = (t & 15) << 4;
        int grow = node0 + row;
        float4* dst4 = reinterpret_cast<float4*>(&Xs[row][col]);
        if (grow < N) {
            const float4* src4 =
                reinterpret_cast<const float4*>(X + (size_t)grow * D + col);
            dst4[0] = src4[0]; dst4[1] = src4[1];
            dst4[2] = src4[2]; dst4[3] = src4[3];
        } else {
            float4 z = make_float4(0.f, 0.f, 0.f, 0.f);
            dst4[0] = z; dst4[1] = z; dst4[2] = z; dst4[3] = z;
        }
    }
    __syncthreads();

    const int lane  = t & 31;
    const int wave  = t >> 5;
    const int m     = lane & 15;          // A row / B col within tile
    const int khalf = (lane >> 4) << 1;   // 0 (lanes 0-15) or 2 (lanes 16-31)
    const int rbase = (lane >> 4) << 3;   // D-matrix row offset: 0 or 8

    for (int i = 0; i < 6; ++i) {
        const int tile = wave * 6 + i;        // 0..47
        const int mat  = tile >> 4;           // 0=U, 1=V, 2=Y
        const int col0 = (tile & 15) << 4;

        const float* Wp;  int wstride; int koff;
        const float* bias; float* Out;
        if (mat == 0)      { Wp = W_in;  wstride = 2 * D; koff = 0; bias = nullptr; Out = U; }
        else if (mat == 1) { Wp = W_in;  wstride = 2 * D; koff = D; bias = b_in;    Out = V; }
        else               { Wp = W_out; wstride = D;     koff = 0; bias = b_out;   Out = Y; }

        const float* wrow = Wp + (size_t)(col0 + m) * wstride + koff;

        v8f acc = {};
        for (int k = 0; k < D; k += 4) {
            const int kk = k + khalf;
            float2 af = *reinterpret_cast<const float2*>(&Xs[m][kk]); // ds_load_b64
            float2 bf = *reinterpret_cast<const float2*>(wrow + kk);  // global b64
            v2f a; a[0] = af.x; a[1] = af.y;
            v2f b; b[0] = bf.x; b[1] = bf.y;
            acc = __builtin_amdgcn_wmma_f32_16x16x4_f32(
                      false, a, false, b, (short)0, acc, false, false);
        }

        const float bv = bias ? bias[col0 + m] : 0.0f;
        #pragma unroll
        for (int r = 0; r < 8; ++r) {
            int grow = node0 + rbase + r;     // D layout: M = r (+8 for hi lanes)
            if (grow < N)
                Out[(size_t)grow * D + col0 + m] = acc[r] + bv;
        }
    }
}

// ---------------------------------------------------------------------------
// Kernel 3: edge pass. One wave32 per edge:
//   logit = a_b + sum_k a_w[k] * lrelu(U[src,k] + V[tgt,k])
//   S[tgt] += exp(logit)
// U/V rows live in L2 (20 MB total); float4 loads are fully coalesced.
// ---------------------------------------------------------------------------
__global__ __launch_bounds__(256)
void gat_edge_kernel(const float* __restrict__ U,
                     const float* __restrict__ V,
                     const int* __restrict__ src,
                     const int* __restrict__ tgt,
                     const float* __restrict__ a_w,
                     const float* __restrict__ a_b,
                     float* __restrict__ S,
                     int E) {
    const int lane = threadIdx.x & 31;
    const int eid  = blockIdx.x * 8 + (threadIdx.x >> 5);
    if (eid >= E) return;

    const int sn = src[eid];
    const int tn = tgt[eid];
    const float4* Ur = reinterpret_cast<const float4*>(U + (size_t)sn * D);
    const float4* Vr = reinterpret_cast<const float4*>(V + (size_t)tn * D);
    const float4* Ar = reinterpret_cast<const float4*>(a_w);

    float sum = 0.0f;
    #pragma unroll
    for (int half = 0; half < 2; ++half) {      // 64 float4s / 32 lanes
        const int i = lane + half * 32;
        float4 u = Ur[i], v = Vr[i], a = Ar[i];
        sum += a.x * lrelu(u.x + v.x);
        sum += a.y * lrelu(u.y + v.y);
        sum += a.z * lrelu(u.z + v.z);
        sum += a.w * lrelu(u.w + v.w);
    }
    #pragma unroll
    for (int off = 16; off > 0; off >>= 1)
        sum += __shfl_down(sum, off, 32);

    if (lane == 0)
        atomicAdd(&S[tn], expf(sum + a_b[0]));
}

// ---------------------------------------------------------------------------
// Kernel 4: out[n, :] = lrelu( Y[n, :] * S[n] / (S[n] + eps) )
// ---------------------------------------------------------------------------
__global__ void gat_final_kernel(const float* __restrict__ Y,
                                 const float* __restrict__ S,
                                 float* __restrict__ out,
                                 int total) {
    int i = blockIdx.x * blockDim.x + threadIdx.x;
    if (i >= total) return;
    float s = S[i >> 8];                 // D == 256
    float v = Y[i] * (s / (s + EPS_F));
    out[i] = lrelu(v);
}

// ---------------------------------------------------------------------------
extern "C" void kernel_launch(void* const* d_in, const int* in_sizes, int n_in,
                              void* d_out, int out_size, void* d_ws, size_t ws_size,
                              hipStream_t stream) {
    const float* x     = (const float*)d_in[0];
    const int*   src   = (const int*)  d_in[1];
    const int*   tgt   = (const int*)  d_in[2];
    const float* W_in  = (const float*)d_in[3];
    const float* b_in  = (const float*)d_in[4];
    const float* a_w   = (const float*)d_in[5];
    const float* a_b   = (const float*)d_in[6];
    const float* W_out = (const float*)d_in[7];
    const float* b_out = (const float*)d_in[8];
    float* out = (float*)d_out;

    const int N = in_sizes[0] / D;
    const int E = in_sizes[1];

    float* U = (float*)d_ws;           // [N, 256]
    float* V = U + (size_t)N * D;      // [N, 256] (includes b_in)
    float* Y = V + (size_t)N * D;      // [N, 256] (includes b_out)
    float* S = Y + (size_t)N * D;      // [N]

    gat_zero_kernel<<<(N + 255) / 256, 256, 0, stream>>>(S, N);

    const int ntiles = (N + 15) / 16;
    gat_node_gemm_kernel<<<ntiles, 256, 0, stream>>>(
        x, W_in, b_in, W_out, b_out, U, V, Y, N);

    gat_edge_kernel<<<(E + 7) / 8, 256, 0, stream>>>(
        U, V, src, tgt, a_w, a_b, S, E);

    const int total = N * D;
    gat_final_kernel<<<(total + 255) / 256, 256, 0, stream>>>(Y, S, out, total);
}